// DiagonalwiseSeparableLayer_1176821039919
// MI455X (gfx1250) — compile-verified
//
#include <hip/hip_runtime.h>

typedef __attribute__((ext_vector_type(2))) float v2f;
typedef __attribute__((ext_vector_type(8))) float v8f;

namespace {
constexpr int B    = 16;
constexpr int H    = 112;
constexpr int W    = 112;
constexpr int Cin  = 256;
constexpr int G    = 8;
constexpr int CG   = 32;          // channels per group
constexpr int Cout = 512;
constexpr int NG   = 64;          // output channels per group
constexpr int XW   = W + 2;       // zero-padded width in LDS
constexpr int YSTR = 34;          // LDS row stride for y (even -> 8B aligned, spreads banks)
constexpr int NTHREADS = 128;     // 4 waves
}

__global__ __launch_bounds__(NTHREADS)
void fused_dw_pw_wmma(const float* __restrict__ x,
                      const float* __restrict__ splitw,
                      const float* __restrict__ pw,
                      float* __restrict__ out)
{
    __shared__ float xl[3 * XW * CG];   // 3*114*32 f32 = 43776 B
    __shared__ float yl[W * YSTR];      // 112*34  f32 = 15232 B

    const int blk  = blockIdx.x;
    const int g    = blk & (G - 1);
    const int bh   = blk >> 3;
    const int h    = bh % H;
    const int b    = bh / H;
    const int tid  = threadIdx.x;
    const int lane = tid & 31;
    const int wv   = tid >> 5;          // wave id 0..3 -> N-tile id

    // ---------------- B-matrix (pw group slice) into registers ----------------
    // wave wv covers output columns [g*64 + wv*16, +16)
    // V_WMMA B 4x16 layout: VGPR0 lanes0-15 K=0 / lanes16-31 K=2; VGPR1 K=1 / K=3
    const int ocol  = g * NG + wv * 16 + (lane & 15);
    const int krow0 = 2 * (lane >> 4);  // 0 or 2
    v2f breg[8];
#pragma unroll
    for (int ks = 0; ks < 8; ++ks) {
        const int k = ks * 4 + krow0;
        breg[ks].x = pw[(size_t)(k + 0) * Cout + ocol];
        breg[ks].y = pw[(size_t)(k + 1) * Cout + ocol];
    }

    // ---------------- stage padded x rows h-1..h+1 into LDS -------------------
    // layout: xl[(r*XW + wp)*CG + ch], actual w = wp - 1; zero for OOB (SAME pad)
    const int total4 = 3 * XW * CG / 4;     // float4 slots = 2736
    for (int idx = tid; idx < total4; idx += NTHREADS) {
        const int ch4  = idx & (CG / 4 - 1);    // 0..7
        const int rest = idx >> 3;              // r*XW + wp
        const int wp   = rest % XW;
        const int r    = rest / XW;
        const int hh   = h - 1 + r;
        const int w    = wp - 1;
        float4 v = make_float4(0.f, 0.f, 0.f, 0.f);
        if (hh >= 0 && hh < H && w >= 0 && w < W) {
            v = *(const float4*)(x + (((size_t)b * H + hh) * W + w) * Cin + g * CG + ch4 * 4);
        }
        *(float4*)(&xl[idx * 4]) = v;
    }
    __syncthreads();

    // ---------------- depthwise 3x3 (diagonal weights) -> y in LDS ------------
    const int ch = tid & (CG - 1);
    const int wq = (tid >> 5) * 28;         // 4 waves x 28 widths = 112
    float dwgt[3][3];
#pragma unroll
    for (int r = 0; r < 3; ++r)
#pragma unroll
        for (int s = 0; s < 3; ++s)
            dwgt[r][s] = splitw[((((size_t)g * 3 + r) * 3 + s) * CG + ch) * CG + ch];

    for (int w = wq; w < wq + 28; ++w) {
        float acc = 0.f;
#pragma unroll
        for (int r = 0; r < 3; ++r)
#pragma unroll
            for (int s = 0; s < 3; ++s)
                acc = fmaf(xl[(r * XW + (w + s)) * CG + ch], dwgt[r][s], acc);
        yl[w * YSTR + ch] = acc;            // y[w][ch], row stride 34
    }
    __syncthreads();

    // ---------------- grouped pointwise GEMM via V_WMMA_F32_16X16X4_F32 -------
    // per wave: 7 M-tiles x (K=32 -> 8 wmma)
    const int mlane = lane & 15;
    const size_t out_row0 = (((size_t)b * H + h) * W) * Cout;
    for (int mt = 0; mt < 7; ++mt) {
        const int m0 = mt * 16;
        v8f acc = {};
#pragma unroll
        for (int ks = 0; ks < 8; ++ks) {
            // A 16x4 layout: lanes0-15 hold K=k,k+1 ; lanes16-31 hold K=k+2,k+3
            const int k = ks * 4 + krow0;
            const v2f a = *(const v2f*)(&yl[(m0 + mlane) * YSTR + k]);   // ds_load_b64
            acc = __builtin_amdgcn_wmma_f32_16x16x4_f32(
                /*neg_a=*/false, a, /*neg_b=*/false, breg[ks],
                /*c_mod=*/(short)0, acc, /*reuse_a=*/false, /*reuse_b=*/false);
        }
        // C/D layout: VGPR r -> M = m0 + r + 8*(lane>=16), N = ocol
#pragma unroll
        for (int r = 0; r < 8; ++r) {
            const int m = m0 + r + 8 * (lane >> 4);
            out[out_row0 + (size_t)m * Cout + ocol] = acc[r];
        }
    }
}

extern "C" void kernel_launch(void* const* d_in, const int* in_sizes, int n_in,
                              void* d_out, int out_size, void* d_ws, size_t ws_size,
                              hipStream_t stream) {
    (void)in_sizes; (void)n_in; (void)d_ws; (void)ws_size; (void)out_size;
    const float* x      = (const float*)d_in[0];
    const float* splitw = (const float*)d_in[1];
    const float* pw     = (const float*)d_in[2];
    float* out          = (float*)d_out;
    const dim3 grid(B * H * G);   // 14336 workgroups: one per (b, h, group)
    fused_dw_pw_wmma<<<grid, NTHREADS, 0, stream>>>(x, splitw, pw, out);
}